// DeformMaxPool2d_22290880266905
// MI455X (gfx1250) — compile-verified
//
#include <hip/hip_runtime.h>

// Problem constants (from reference): B=16, C=64, DIM=256, OUT=128, K*K=4
#define DIM      256
#define OUTD     128
#define IMG      (DIM * DIM)       // 65536 elements = 256 KB per image
#define OPIX     (OUTD * OUTD)     // 16384 output pixels per image
#define NIMG     (16 * 64)         // 1024 images
#define NTHREADS 256               // 8 wave32 waves
#define CHUNK    (NTHREADS * 16)   // 4096 bytes copied per block step
#define NSTEP    (IMG * 4 / CHUNK) // 64 async b128 loads per lane

// One async global->LDS 16B load with a compile-time immediate offset.
// ISA: the immediate offset is added to both the LDS and global address (GVS mode),
// so a single (loff, goff) VGPR pair serves the whole copy.
template <int OFF>
__device__ __forceinline__ void async_b128(uint32_t loff, uint32_t goff,
                                           const float* src) {
    asm volatile("global_load_async_to_lds_b128 %0, %1, %2 offset:%c3"
                 :
                 : "v"(loff), "v"(goff), "s"(src), "i"(OFF)
                 : "memory");
}

template <int K, int N>
struct CopyUnroll {
    __device__ static __forceinline__ void run(uint32_t loff, uint32_t goff,
                                               const float* src) {
        async_b128<K * CHUNK>(loff, goff, src);
        CopyUnroll<K + 1, N>::run(loff, goff, src);
    }
};
template <int N>
struct CopyUnroll<N, N> {
    __device__ static __forceinline__ void run(uint32_t, uint32_t, const float*) {}
};

__global__ __launch_bounds__(NTHREADS)
void deform_maxpool_async_lds(const float* __restrict__ x,
                              const int*   __restrict__ idx,
                              float* __restrict__ out) {
    extern __shared__ float lds[];   // 256 KB dynamic LDS: one full image

    const int tid = threadIdx.x;
    const int img = blockIdx.x;                       // b*C + c
    const float* __restrict__ src = x + (size_t)img * IMG;
    float* __restrict__ dst       = out + (size_t)img * OPIX;

    // ---- Phase 1: async copy the whole 256 KB image into LDS ----
    // Fixed per-lane base addresses; all 64 steps use immediate offsets only.
    const uint32_t goff = (uint32_t)tid * 16u;
    const uint32_t loff = (uint32_t)(uintptr_t)lds + goff;
    CopyUnroll<0, NSTEP>::run(loff, goff, src);

    // Each wave drains its own ASYNCcnt, then a workgroup barrier makes the
    // LDS image visible to all 8 waves.
#if __has_builtin(__builtin_amdgcn_s_wait_asynccnt)
    __builtin_amdgcn_s_wait_asynccnt(0);
#else
    asm volatile("s_wait_asynccnt 0" ::: "memory");
#endif
    __syncthreads();

    // ---- Phase 2: gather 4 pixels per output from LDS, max-reduce ----
    const int4* __restrict__ iv4 = (const int4*)idx;  // (OUT*OUT) int4 entries
    #pragma unroll 4
    for (int k = 0; k < OPIX / NTHREADS; ++k) {
        int p = tid + k * NTHREADS;                   // coalesced idx + out
        int4 iv = iv4[p];
        float a = lds[iv.x];
        float b = lds[iv.y];
        float c = lds[iv.z];
        float d = lds[iv.w];
        dst[p] = fmaxf(fmaxf(a, b), fmaxf(c, d));
    }
}

extern "C" void kernel_launch(void* const* d_in, const int* in_sizes, int n_in,
                              void* d_out, int out_size, void* d_ws, size_t ws_size,
                              hipStream_t stream) {
    const float* x   = (const float*)d_in[0];   // (16,64,256,256) f32
    const int*   idx = (const int*)d_in[1];     // (128,128,4) i32
    float*       out = (float*)d_out;           // (16,64,128,128) f32

    dim3 grid(NIMG);
    dim3 block(NTHREADS);
    size_t lds_bytes = (size_t)IMG * sizeof(float);  // 262144 B (<= 320KB/WGP)
    deform_maxpool_async_lds<<<grid, block, lds_bytes, stream>>>(x, idx, out);
}